// NeuralSFA_61254823575597
// MI455X (gfx1250) — compile-verified
//
#include <hip/hip_runtime.h>
#include <math.h>

// ---------------------------------------------------------------------------
// NeuralSFA fused pipeline for gfx1250 (MI455X).
// Kernel 1: one workgroup (128 thr / 4 waves) per image. Entire CNN in LDS.
//   conv1: VALU direct conv, weights register-resident, 4x4 patch loads.
//   conv2: im2col panel in LDS, GEMM 128x16x72  via V_WMMA_F32_16X16X4_F32
//   conv3: im2col panel in LDS, GEMM 16x32x144  via V_WMMA_F32_16X16X4_F32
//   Result panels padded so WMMA writeback is unconditional ds_store_b128.
// Kernel 2: transition-matrix softmax + 32-step automaton scan (1 thread / b).
// ---------------------------------------------------------------------------

typedef __attribute__((ext_vector_type(2))) float v2f;
typedef __attribute__((ext_vector_type(4))) float v4f;
typedef __attribute__((ext_vector_type(8))) float v8f;

#if defined(__HIP_DEVICE_COMPILE__) && !__has_builtin(__builtin_amdgcn_wmma_f32_16x16x4_f32)
#error "__builtin_amdgcn_wmma_f32_16x16x4_f32 not available on this toolchain"
#endif

__device__ __forceinline__ v8f wmma_f32_k4(v2f a, v2f b, v8f c) {
    return __builtin_amdgcn_wmma_f32_16x16x4_f32(false, a, false, b, (short)0, c,
                                                 false, false);
}

__global__ __launch_bounds__(128)
void neural_sfa_cnn_kernel(const float* __restrict__ seq,
                           const float* __restrict__ c1w, const float* __restrict__ c1b,
                           const float* __restrict__ c2w, const float* __restrict__ c2b,
                           const float* __restrict__ c3w, const float* __restrict__ c3b,
                           const float* __restrict__ dw,  const float* __restrict__ db,
                           float* __restrict__ preds,     // (32,256,10)
                           float* __restrict__ guards)    // (32,256,6)
{
    __shared__ __align__(16) float img[784];      // 1x28x28
    __shared__ __align__(16) float w1[72];        // (8,1,3,3)
    __shared__ __align__(16) float b1[8];
    __shared__ __align__(16) float w2[1152];      // (16,8,3,3) -> [n][k]
    __shared__ __align__(16) float b2[16];
    __shared__ __align__(16) float w3[4608];      // (32,16,3,3) -> [ch][k]
    __shared__ __align__(16) float b3[32];
    __shared__ __align__(16) float s1[1352];      // conv1 pooled: [8][13][13]
    __shared__ __align__(16) float A2[128 * 72];  // conv2 im2col panel (rows 121..127 = 0)
    __shared__ __align__(16) float c2r[16 * 128]; // conv2 relu panel, padded cols (121..127 junk)
    __shared__ __align__(16) float s2[400];       // conv2 pooled: [16][5][5]
    __shared__ __align__(16) float A3[16 * 144];  // conv3 im2col panel (rows 9..15 = 0)
    __shared__ __align__(16) float c3r[32 * 16];  // conv3 relu panel, padded cols (9..15 junk)
    __shared__ __align__(16) float s3[32];        // conv3 pooled (flat features)
    __shared__ __align__(16) float lg[10];
    __shared__ __align__(16) float pr[10];

    const int tid  = threadIdx.x;
    const int lane = tid & 31;
    const int wv   = tid >> 5;
    const int img_idx = blockIdx.x;        // memory order: b*32 + n
    const int b = img_idx >> 5;
    const int n = img_idx & 31;

    // ---- stage image + weights into LDS (each input byte read exactly once) ----
    const float* src = seq + (size_t)img_idx * 784;
    for (int i = tid; i < 784; i += 128)  img[i] = src[i];
    for (int i = tid; i < 72; i += 128)   w1[i] = c1w[i];
    if (tid < 8)  b1[tid] = c1b[tid];
    for (int i = tid; i < 1152; i += 128) w2[i] = c2w[i];
    if (tid < 16) b2[tid] = c2b[tid];
    for (int i = tid; i < 4608; i += 128) w3[i] = c3w[i];
    if (tid < 32) b3[tid] = c3b[tid];
    __syncthreads();

    // ---- conv1 (1->8, 3x3) + relu + avgpool2 -> [8][13][13] (VALU) ----
    // Thread owns channel (tid&7): weights + bias register-resident; each pooled
    // output needs one 4x4 input patch (16 LDS loads) computed from registers.
    {
        const int c = tid & 7;
        float wr[9];
        #pragma unroll
        for (int k = 0; k < 9; ++k) wr[k] = w1[c * 9 + k];
        const float bias = b1[c];
        for (int p = tid >> 3; p < 169; p += 16) {
            const int py = p / 13, px = p - py * 13;
            const float* ib = &img[(2 * py) * 28 + 2 * px];
            float patch[16];
            #pragma unroll
            for (int r = 0; r < 4; ++r)
                #pragma unroll
                for (int q = 0; q < 4; ++q)
                    patch[r * 4 + q] = ib[r * 28 + q];
            float acc = 0.f;
            #pragma unroll
            for (int dy = 0; dy < 2; ++dy) {
                #pragma unroll
                for (int dx = 0; dx < 2; ++dx) {
                    float v = bias;
                    #pragma unroll
                    for (int ky = 0; ky < 3; ++ky)
                        #pragma unroll
                        for (int kx = 0; kx < 3; ++kx)
                            v = fmaf(patch[(dy + ky) * 4 + dx + kx], wr[ky * 3 + kx], v);
                    acc += fmaxf(v, 0.f);
                }
            }
            s1[c * 169 + p] = 0.25f * acc;
        }
    }
    __syncthreads();

    // ---- conv2 im2col panel: A2[m][k], m = oy*11+ox (128 rows, pad with 0) ----
    {
        const int m = tid;                         // one row per thread
        float* dst = &A2[m * 72];
        if (m < 121) {
            const int moy = m / 11, mox = m - moy * 11;
            const float* sb = &s1[moy * 13 + mox];
            #pragma unroll
            for (int c = 0; c < 8; ++c)
                #pragma unroll
                for (int ky = 0; ky < 3; ++ky)
                    #pragma unroll
                    for (int kx = 0; kx < 3; ++kx)
                        dst[c * 9 + ky * 3 + kx] = sb[c * 169 + ky * 13 + kx];
        } else {
            #pragma unroll
            for (int k = 0; k < 72; ++k) dst[k] = 0.f;
        }
    }
    __syncthreads();

    // ---- conv2 GEMM: [128x72] x [72x16] in fp32 WMMA 16x16x4 tiles ----
    {
        const int halfhi = (lane >= 16) ? 1 : 0;
        const int ncol   = lane & 15;
        const int mrow   = lane & 15;
        const float bias = b2[ncol];
        // hoist B fragments (reused by both M-tiles of this wave): 18 x v2f
        const float* brow = &w2[ncol * 72];
        v2f bf[18];
        #pragma unroll
        for (int s = 0; s < 18; ++s)
            bf[s] = *(const v2f*)&brow[4 * s + 2 * halfhi];

        #pragma unroll
        for (int t = 0; t < 2; ++t) {                // straight-line, no loop exec dance
            const int tile = wv + 4 * t;
            const float* arow = &A2[(tile * 16 + mrow) * 72];
            v8f acc = {};
            #pragma unroll
            for (int s = 0; s < 18; ++s) {
                v2f a = *(const v2f*)&arow[4 * s + 2 * halfhi];
                acc = wmma_f32_k4(a, bf[s], acc);
            }
            // D: VGPR v -> M = tile*16 + v (+8 upper half-wave), N = ncol.
            // Padded panel -> unconditional vectorized stores.
            v4f lo, hi;
            #pragma unroll
            for (int v = 0; v < 4; ++v) lo[v] = fmaxf(acc[v] + bias, 0.f);
            #pragma unroll
            for (int v = 0; v < 4; ++v) hi[v] = fmaxf(acc[4 + v] + bias, 0.f);
            float* drow = &c2r[ncol * 128 + tile * 16 + 8 * halfhi];
            *(v4f*)&drow[0] = lo;
            *(v4f*)&drow[4] = hi;
        }
    }
    __syncthreads();

    // ---- avgpool2 conv2 -> [16][5][5] ----
    for (int idx = tid; idx < 400; idx += 128) {
        int c = idx / 25, rem = idx - c * 25;
        int py = rem / 5, px = rem - py * 5;
        const float* base = &c2r[c * 128];
        s2[idx] = 0.25f * (base[(2 * py) * 11 + 2 * px]     + base[(2 * py) * 11 + 2 * px + 1] +
                           base[(2 * py + 1) * 11 + 2 * px] + base[(2 * py + 1) * 11 + 2 * px + 1]);
    }
    __syncthreads();

    // ---- conv3 im2col panel: A3[row][k], row = oy*3+ox (16 rows, pad with 0) ----
    {
        const int row   = tid >> 3;                 // 0..15
        const int chunk = tid & 7;                  // k = chunk*18 .. +17 (18 = 2*9)
        float* dst = &A3[row * 144 + chunk * 18];
        if (row < 9) {
            const int moy = row / 3, mox = row - moy * 3;
            const float* sb = &s2[chunk * 50 + moy * 5 + mox];
            #pragma unroll
            for (int j = 0; j < 18; ++j) {
                const int c  = j / 9;               // 0 or 1 within chunk
                const int kk = j - c * 9;
                const int ky = kk / 3, kx = kk - ky * 3;
                dst[j] = sb[c * 25 + ky * 5 + kx];
            }
        } else {
            #pragma unroll
            for (int j = 0; j < 18; ++j) dst[j] = 0.f;
        }
    }
    __syncthreads();

    // ---- conv3 GEMM: [16x144] x [144x32], two N-tiles on waves 0/1 ----
    if (wv < 2) {                                    // wave-uniform branch
        const int halfhi = (lane >= 16) ? 1 : 0;
        const int ncol   = lane & 15;
        const int ch     = wv * 16 + ncol;
        const float bias = b3[ch];
        const float* arow = &A3[(lane & 15) * 144];
        const float* brow = &w3[ch * 144];
        v8f acc = {};
        #pragma unroll
        for (int s = 0; s < 36; ++s) {
            v2f a   = *(const v2f*)&arow[4 * s + 2 * halfhi];
            v2f bfr = *(const v2f*)&brow[4 * s + 2 * halfhi];
            acc = wmma_f32_k4(a, bfr, acc);
        }
        v4f lo, hi;
        #pragma unroll
        for (int v = 0; v < 4; ++v) lo[v] = fmaxf(acc[v] + bias, 0.f);
        #pragma unroll
        for (int v = 0; v < 4; ++v) hi[v] = fmaxf(acc[4 + v] + bias, 0.f);
        float* drow = &c3r[ch * 16 + 8 * halfhi];
        *(v4f*)&drow[0] = lo;
        *(v4f*)&drow[4] = hi;
    }
    __syncthreads();

    // ---- avgpool2 of 3x3 (floor) keeps window {0,1,3,4} -> 32 features ----
    if (tid < 32) {
        const float* base = &c3r[tid * 16];
        s3[tid] = 0.25f * (base[0] + base[1] + base[3] + base[4]);
    }
    __syncthreads();

    // ---- dense (32->10) + softmax ----
    if (tid < 10) {
        float v = db[tid];
        #pragma unroll
        for (int c = 0; c < 32; ++c) v = fmaf(s3[c], dw[tid * 32 + c], v);
        lg[tid] = v;
    }
    __syncthreads();
    if (tid == 0) {
        float mx = lg[0];
        #pragma unroll
        for (int j = 1; j < 10; ++j) mx = fmaxf(mx, lg[j]);
        float e[10], sum = 0.f;
        #pragma unroll
        for (int j = 0; j < 10; ++j) { e[j] = __expf(lg[j] - mx); sum += e[j]; }
        float inv = 1.f / sum;
        #pragma unroll
        for (int j = 0; j < 10; ++j) pr[j] = e[j] * inv;
    }
    __syncthreads();

    // ---- emit digit probs (N,B,10) and guard probs (N,B,6) ----
    const size_t obase = (size_t)n * 256 + b;
    if (tid < 10) preds[obase * 10 + tid] = pr[tid];
    if (tid == 0) {
        float* g = guards + obase * 6;
        g[0] = pr[8];
        g[1] = pr[4] + pr[6];
        g[2] = pr[0] + pr[2];
        g[3] = pr[7] + pr[9];
        g[4] = pr[5];
        g[5] = pr[1] + pr[3];
    }
}

// ---------------------------------------------------------------------------
// Kernel 2: build transition matrices M[6][10][10] (softmax(trans/0.1) + fixed
// accepting row), then run the 32-step scan; one thread per batch element.
// ---------------------------------------------------------------------------
__global__ __launch_bounds__(256)
void neural_sfa_scan_kernel(const float* __restrict__ trans,   // (6,9,10)
                            const float* __restrict__ guards,  // (32,256,6)
                            float* __restrict__ state_out)     // (256,10)
{
    __shared__ float M[600]; // [r][i][j]
    const int tid = threadIdx.x;

    if (tid < 54) {
        int r = tid / 9, i = tid - r * 9;
        const float* row = trans + (r * 9 + i) * 10;
        float sc[10];
        float mx = -1e30f;
        #pragma unroll
        for (int j = 0; j < 10; ++j) { sc[j] = row[j] * 10.f; mx = fmaxf(mx, sc[j]); } // /TEMP, TEMP=0.1
        float sum = 0.f;
        #pragma unroll
        for (int j = 0; j < 10; ++j) { sc[j] = __expf(sc[j] - mx); sum += sc[j]; }
        float inv = 1.f / sum;
        #pragma unroll
        for (int j = 0; j < 10; ++j) M[(r * 10 + i) * 10 + j] = sc[j] * inv;
    } else if (tid < 60) {
        int r = tid - 54;
        #pragma unroll
        for (int j = 0; j < 10; ++j) M[(r * 10 + 9) * 10 + j] = (j == 9) ? 1.f : 0.f;
    }
    __syncthreads();

    const int b = tid; // 256 batch elements
    float st[10];
    #pragma unroll
    for (int j = 0; j < 10; ++j) st[j] = (j == 0) ? 1.f : 0.f;

    for (int n = 0; n < 32; ++n) {
        const float* g = guards + ((size_t)n * 256 + b) * 6;
        float gr[6];
        #pragma unroll
        for (int r = 0; r < 6; ++r) gr[r] = g[r];
        float ns[10];
        #pragma unroll
        for (int j = 0; j < 10; ++j) {
            float acc = 0.f;
            #pragma unroll
            for (int i = 0; i < 10; ++i) {
                float aij = 0.f;
                #pragma unroll
                for (int r = 0; r < 6; ++r) aij = fmaf(gr[r], M[(r * 10 + i) * 10 + j], aij);
                acc = fmaf(st[i], aij, acc);
            }
            ns[j] = acc;
        }
        #pragma unroll
        for (int j = 0; j < 10; ++j) st[j] = ns[j];
    }
    float* out = state_out + b * 10;
    #pragma unroll
    for (int j = 0; j < 10; ++j) out[j] = st[j];
}

extern "C" void kernel_launch(void* const* d_in, const int* in_sizes, int n_in,
                              void* d_out, int out_size, void* d_ws, size_t ws_size,
                              hipStream_t stream) {
    (void)in_sizes; (void)n_in; (void)d_ws; (void)ws_size; (void)out_size;
    const float* seq   = (const float*)d_in[0];
    const float* c1w   = (const float*)d_in[1];
    const float* c1b   = (const float*)d_in[2];
    const float* c2w   = (const float*)d_in[3];
    const float* c2b   = (const float*)d_in[4];
    const float* c3w   = (const float*)d_in[5];
    const float* c3b   = (const float*)d_in[6];
    const float* dw    = (const float*)d_in[7];
    const float* db    = (const float*)d_in[8];
    const float* trans = (const float*)d_in[9];

    float* out    = (float*)d_out;
    float* preds  = out;                 // 32*256*10 = 81920
    float* guards = out + 81920;         // 32*256*6  = 49152
    float* statef = out + 131072;        // 256*10    = 2560

    neural_sfa_cnn_kernel<<<8192, 128, 0, stream>>>(seq, c1w, c1b, c2w, c2b,
                                                    c3w, c3b, dw, db, preds, guards);
    neural_sfa_scan_kernel<<<1, 256, 0, stream>>>(trans, guards, statef);
}